// MatchaTTS_62998580297791
// MI455X (gfx1250) — compile-verified
//
#include <hip/hip_runtime.h>
#include <math.h>

#define B_  32
#define TX_ 512
#define TY_ 2048
#define NF_ 80
#define NEGV (-1e9f)
#define LOG2PI_F 1.8378770664093453f
#define CONST_TERM (-0.5f * LOG2PI_F * (float)NF_)

typedef __attribute__((ext_vector_type(2))) float v2f;
typedef __attribute__((ext_vector_type(8))) float v8f;

// ---------------------------------------------------------------------------
// Kernel 1: per-position squared sums: out[b,i] = sum_c src[b,c,i]^2
// ---------------------------------------------------------------------------
__global__ void sumsq_k(const float* __restrict__ src, float* __restrict__ out, int L) {
    int g = blockIdx.x * blockDim.x + threadIdx.x;
    if (g >= B_ * L) return;
    int b = g / L, i = g - b * L;
    const float* s = src + (size_t)b * NF_ * L + i;
    float acc = 0.f;
#pragma unroll 8
    for (int c = 0; c < NF_; ++c) { float v = s[(size_t)c * L]; acc += v * v; }
    out[g] = acc;
}

// ---------------------------------------------------------------------------
// Kernel 2: WMMA f32 GEMM + log-prior, stored transposed as lp[b, y, x]
//   lp = const - 0.5*musq[x] - 0.5*ysq[y] + sum_c mu[b,c,x]*y[b,c,y], masked.
// One wave computes a 16x16 (x,y) tile via V_WMMA_F32_16X16X4_F32 (K=80 -> 20).
// ---------------------------------------------------------------------------
__global__ void __launch_bounds__(128)
gemm_logprior_k(const float* __restrict__ mu, const float* __restrict__ y,
                const float* __restrict__ musq, const float* __restrict__ ysq,
                const int* __restrict__ xlen, const int* __restrict__ ylen,
                float* __restrict__ lp) {
    const int lane = threadIdx.x & 31;
    const int wv   = threadIdx.x >> 5;
    const int b    = blockIdx.z;
    const int x0   = blockIdx.x * 16;
    const int y0   = (blockIdx.y * 4 + wv) * 16;
    const int m    = lane & 15;
    const int hi   = lane >> 4;              // 0 for lanes 0-15, 1 for 16-31

    const float* muB = mu + (size_t)b * NF_ * TX_;
    const float* yB  = y  + (size_t)b * NF_ * TY_;

    v8f acc = {};
#pragma unroll
    for (int k0 = 0; k0 < NF_; k0 += 4) {
        // A (16x4 f32): lanes 0-15 hold K=k0,k0+1 ; lanes 16-31 hold K=k0+2,k0+3
        int ka = k0 + 2 * hi;
        v2f a, bb;
        a.x  = muB[(size_t)ka       * TX_ + x0 + m];
        a.y  = muB[(size_t)(ka + 1) * TX_ + x0 + m];
        // B (4x16 f32): same K split, N striped across lanes
        bb.x = yB[(size_t)ka       * TY_ + y0 + m];
        bb.y = yB[(size_t)(ka + 1) * TY_ + y0 + m];
        acc = __builtin_amdgcn_wmma_f32_16x16x4_f32(
            false, a, false, bb, (short)0, acc, false, false);
    }

    const int xl = xlen[b], yl = ylen[b];
    float* lpB = lp + (size_t)b * TY_ * TX_;
#pragma unroll
    for (int r = 0; r < 8; ++r) {
        int M = r + 8 * hi;      // D layout: VGPR r -> M=r (lanes 0-15), M=r+8 (16-31)
        int N = m;
        int gx = x0 + M, gy = y0 + N;
        float val = acc[r] + CONST_TERM
                  - 0.5f * (musq[b * TX_ + gx] + ysq[b * TY_ + gy]);
        if (gx >= xl || gy >= yl) val = NEGV;
        lpB[(size_t)gy * TX_ + gx] = val;
    }
}

// ---------------------------------------------------------------------------
// Kernel 3: Viterbi forward. One wave32 per batch; lane owns 16 consecutive i.
// No barriers: neighbor dependency via register shift + __shfl_up boundary.
// dirs[b, j] = 512-bit mask (16 u32 words), bit i = "came from diagonal".
// ---------------------------------------------------------------------------
__global__ void __launch_bounds__(32)
dp_forward_k(const float* __restrict__ lp, unsigned* __restrict__ dirs) {
    const int b    = blockIdx.x;
    const int lane = threadIdx.x;                  // 0..31
    const float* lpB = lp + (size_t)b * TY_ * TX_;
    unsigned* dirB   = dirs + (size_t)b * TY_ * (TX_ / 32);

    float v[16];
#pragma unroll
    for (int k = 0; k < 16; ++k) v[k] = NEGV;
    if (lane == 0) v[0] = lpB[0];                  // j=0: only i==0 live
    if (lane < TX_ / 32) dirB[lane] = 0u;          // dirs[0] = 0

    // preload row j=1 (lane covers i = lane*16 .. lane*16+15)
    float4 cur[4];
    {
        const float4* rp = (const float4*)(lpB + TX_);
#pragma unroll
        for (int q = 0; q < 4; ++q) cur[q] = rp[lane * 4 + q];
    }

    for (int j = 1; j < TY_; ++j) {
        float4 nxt[4];
        if (j + 1 < TY_) {
            const float4* rp = (const float4*)(lpB + (size_t)(j + 1) * TX_);
#pragma unroll
            for (int q = 0; q < 4; ++q) nxt[q] = rp[lane * 4 + q];
        }

        float in = __shfl_up(v[15], 1, 32);        // previous lane's top element
        if (lane == 0) in = NEGV;

        float lprow[16];
        lprow[ 0]=cur[0].x; lprow[ 1]=cur[0].y; lprow[ 2]=cur[0].z; lprow[ 3]=cur[0].w;
        lprow[ 4]=cur[1].x; lprow[ 5]=cur[1].y; lprow[ 6]=cur[1].z; lprow[ 7]=cur[1].w;
        lprow[ 8]=cur[2].x; lprow[ 9]=cur[2].y; lprow[10]=cur[2].z; lprow[11]=cur[2].w;
        lprow[12]=cur[3].x; lprow[13]=cur[3].y; lprow[14]=cur[3].z; lprow[15]=cur[3].w;

        unsigned m16 = 0u;
        float prev = in;
#pragma unroll
        for (int k = 0; k < 16; ++k) {
            float vk = v[k];
            if (prev >= vk) m16 |= (1u << k);      // from_diag (ties -> diagonal)
            v[k] = lprow[k] + fmaxf(vk, prev);
            prev = vk;
        }

        // pack pair of lanes -> one u32 word (even lane low 16, odd lane high 16)
        unsigned hi16 = __shfl_down(m16, 1, 32);
        if ((lane & 1) == 0)
            dirB[(size_t)j * (TX_ / 32) + (lane >> 1)] = (m16 & 0xFFFFu) | (hi16 << 16);

#pragma unroll
        for (int q = 0; q < 4; ++q) cur[q] = nxt[q];
    }
}

// ---------------------------------------------------------------------------
// Kernel 4: backtrack, one wave per batch. Exploits |idx step| <= 1: over a
// 32-row block all bits lie in <= 2 dwords/row. Lane m loads the 2 candidate
// words of row (jhi - m) in one coalesced round; all lanes then redundantly
// walk the 32-step chain in registers via __shfl (uniform control flow).
// Turns 2048 dependent L2 loads into 64 dependent load-rounds.
// ---------------------------------------------------------------------------
__global__ void __launch_bounds__(32)
backtrack_k(const unsigned* __restrict__ dirs,
            const int* __restrict__ xlen, const int* __restrict__ ylen,
            int* __restrict__ idx_sel) {
    const int b    = blockIdx.x;
    const int lane = threadIdx.x;
    const unsigned* dirB = dirs + (size_t)b * TY_ * (TX_ / 32);
    int* selB = idx_sel + (size_t)b * TY_;
    const int yl = ylen[b];
    int idx = xlen[b] - 1;                         // identical on every lane

    for (int jhi = TY_ - 1; jhi >= 0; jhi -= 32) {
        const int j = jhi - lane;                  // this lane's row (>= 0)
        const int wc  = idx >> 5;                  // idx stays in words {wc, wc-1}
        const int wc1 = (wc > 0) ? wc - 1 : 0;
        const unsigned wA = dirB[(size_t)j * (TX_ / 32) + wc];
        const unsigned wB = dirB[(size_t)j * (TX_ / 32) + wc1];

        int mysel = -1;
#pragma unroll
        for (int m = 0; m < 32; ++m) {
            const int jm = jhi - m;                // uniform across lanes
            const unsigned a  = __shfl(wA, m, 32);
            const unsigned bb = __shfl(wB, m, 32);
            if (jm < yl) {
                if (m == lane) mysel = idx;
                const unsigned word = ((idx >> 5) == wc) ? a : bb;
                idx -= (int)((word >> (idx & 31)) & 1u);
                if (idx < 0) idx = 0;
            }
        }
        selB[j] = mysel;
    }
}

// ---------------------------------------------------------------------------
// Kernel 5: duration counts via integer atomics (deterministic).
// ---------------------------------------------------------------------------
__global__ void dur_count_k(const int* __restrict__ idx_sel, int* __restrict__ dur) {
    int g = blockIdx.x * blockDim.x + threadIdx.x;     // over B*TY
    if (g >= B_ * TY_) return;
    int b = g / TY_;
    int s = idx_sel[g];
    if (s >= 0) atomicAdd(&dur[b * TX_ + s], 1);
}

// ---------------------------------------------------------------------------
// Kernel 6: attn[b,x,y] = (idx_sel[b,y] == x); float2 stores, coalesced on y.
// ---------------------------------------------------------------------------
__global__ void attn_write_k(const int* __restrict__ idx_sel, float* __restrict__ attn) {
    size_t g = (size_t)blockIdx.x * blockDim.x + threadIdx.x;  // over B*TX*TY/2
    if (g >= (size_t)B_ * TX_ * TY_ / 2) return;
    int y2 = (int)(g % (TY_ / 2));
    size_t t = g / (TY_ / 2);
    int x = (int)(t % TX_);
    int b = (int)(t / TX_);
    const int2 s = *(const int2*)(idx_sel + (size_t)b * TY_ + 2 * y2);
    float2 o;
    o.x = (s.x == x) ? 1.0f : 0.0f;
    o.y = (s.y == x) ? 1.0f : 0.0f;
    *(float2*)(attn + 2 * g) = o;
}

// ---------------------------------------------------------------------------
// Deterministic fixed-order block reduction
// ---------------------------------------------------------------------------
__device__ float block_reduce_sum(float v) {
    __shared__ float s[256];
    int t = threadIdx.x;
    s[t] = v;
    __syncthreads();
    for (int o = 128; o > 0; o >>= 1) {
        if (t < o) s[t] += s[t + o];
        __syncthreads();
    }
    return s[0];
}

// Kernel 7: duration-loss partial sums (64 blocks x 256 threads over B*TX)
__global__ void __launch_bounds__(256)
dur_loss_k(const float* __restrict__ logw, const int* __restrict__ dur,
           const int* __restrict__ xlen, float* __restrict__ partial) {
    int g = blockIdx.x * blockDim.x + threadIdx.x;
    int b = g / TX_, x = g - b * TX_;
    float val = 0.f;
    if (x < xlen[b]) {
        float d = logw[g] - logf(1e-8f + (float)dur[g]);
        val = d * d;
    }
    float s = block_reduce_sum(val);
    if (threadIdx.x == 0) partial[blockIdx.x] = s;
}

// Kernel 8: prior-loss partial sums (256 blocks x 256 threads over B*TY)
__global__ void __launch_bounds__(256)
prior_loss_k(const float* __restrict__ mu, const float* __restrict__ y,
             const int* __restrict__ idx_sel, const int* __restrict__ ylen,
             float* __restrict__ partial) {
    int g = blockIdx.x * blockDim.x + threadIdx.x;
    int b = g / TY_, yy = g - b * TY_;
    float s = 0.f;
    if (yy < ylen[b]) {
        int xi = idx_sel[g];
        const float* muB = mu + (size_t)b * NF_ * TX_ + xi;
        const float* yB  = y  + (size_t)b * NF_ * TY_ + yy;
#pragma unroll 8
        for (int c = 0; c < NF_; ++c) {
            float d = yB[(size_t)c * TY_] - muB[(size_t)c * TX_];
            s += 0.5f * (d * d + LOG2PI_F);
        }
    }
    float r = block_reduce_sum(s);
    if (threadIdx.x == 0) partial[blockIdx.x] = r;
}

// Kernel 9: finalize scalars (deterministic serial sums)
__global__ void finalize_k(const float* __restrict__ pd, const float* __restrict__ pp,
                           const int* __restrict__ xlen, const int* __restrict__ ylen,
                           float* __restrict__ out) {
    float sd = 0.f, sp = 0.f;
    for (int i = 0; i < 64;  ++i) sd += pd[i];
    for (int i = 0; i < 256; ++i) sp += pp[i];
    int sx = 0, sy = 0;
    for (int b = 0; b < B_; ++b) { sx += xlen[b]; sy += ylen[b]; }
    out[0] = sd / (float)sx;
    out[1] = sp / ((float)sy * (float)NF_);
}

// ---------------------------------------------------------------------------
// Host-side launch
// ---------------------------------------------------------------------------
extern "C" void kernel_launch(void* const* d_in, const int* in_sizes, int n_in,
                              void* d_out, int out_size, void* d_ws, size_t ws_size,
                              hipStream_t stream) {
    const float* mu   = (const float*)d_in[0];   // [B,NF,TX]
    const float* logw = (const float*)d_in[1];   // [B,1,TX]
    const float* y    = (const float*)d_in[2];   // [B,NF,TY]
    const int*   xlen = (const int*)d_in[3];     // [B]
    const int*   ylen = (const int*)d_in[4];     // [B]
    float* out = (float*)d_out;                  // [dur_loss, prior_loss, attn...]

    char* ws = (char*)d_ws;
    const size_t LP_BYTES   = (size_t)B_ * TY_ * TX_ * 4;        // 134 MB
    const size_t DIR_BYTES  = (size_t)B_ * TY_ * (TX_ / 32) * 4; // 4 MB
    const size_t SEL_BYTES  = (size_t)B_ * TY_ * 4;
    const size_t DUR_BYTES  = (size_t)B_ * TX_ * 4;
    const size_t MSQ_BYTES  = (size_t)B_ * TX_ * 4;
    const size_t YSQ_BYTES  = (size_t)B_ * TY_ * 4;

    float*    lp      = (float*)ws;                              ws += LP_BYTES;
    unsigned* dirs    = (unsigned*)ws;                           ws += DIR_BYTES;
    int*      idx_sel = (int*)ws;                                ws += SEL_BYTES;
    int*      dur     = (int*)ws;                                ws += DUR_BYTES;
    float*    musq    = (float*)ws;                              ws += MSQ_BYTES;
    float*    ysq     = (float*)ws;                              ws += YSQ_BYTES;
    float*    pd      = (float*)ws;                              ws += 64 * 4;
    float*    pp      = (float*)ws;

    hipMemsetAsync(dur, 0, DUR_BYTES, stream);

    sumsq_k<<<(B_ * TX_ + 255) / 256, 256, 0, stream>>>(mu, musq, TX_);
    sumsq_k<<<(B_ * TY_ + 255) / 256, 256, 0, stream>>>(y,  ysq,  TY_);

    gemm_logprior_k<<<dim3(TX_ / 16, TY_ / 64, B_), 128, 0, stream>>>(
        mu, y, musq, ysq, xlen, ylen, lp);

    dp_forward_k<<<B_, 32, 0, stream>>>(lp, dirs);
    backtrack_k<<<B_, 32, 0, stream>>>(dirs, xlen, ylen, idx_sel);

    dur_count_k<<<(B_ * TY_ + 255) / 256, 256, 0, stream>>>(idx_sel, dur);

    attn_write_k<<<(unsigned)(((size_t)B_ * TX_ * TY_ / 2 + 255) / 256), 256, 0, stream>>>(
        idx_sel, out + 2);

    dur_loss_k<<<(B_ * TX_) / 256, 256, 0, stream>>>(logw, dur, xlen, pd);
    prior_loss_k<<<(B_ * TY_) / 256, 256, 0, stream>>>(mu, y, idx_sel, ylen, pp);
    finalize_k<<<1, 1, 0, stream>>>(pd, pp, xlen, ylen, out);
}